// MCA_23227183137075
// MI455X (gfx1250) — compile-verified
//
#include <hip/hip_runtime.h>
#include <hip/hip_bf16.h>

typedef __attribute__((ext_vector_type(2))) float v2f;
typedef __attribute__((ext_vector_type(8))) float v8f;

#define Bn 8
#define Cn 64
#define Ln 2048
#define Lh 1024
#define Gn 4
#define Hn 8
#define EPSc 1e-5f

__device__ __forceinline__ v8f wmma4(v2f a, v2f b, v8f c) {
  // V_WMMA_F32_16X16X4_F32 : D(16x16 f32) = A(16x4 f32) * B(4x16 f32) + C
  return __builtin_amdgcn_wmma_f32_16x16x4_f32(
      /*neg_a=*/false, a, /*neg_b=*/false, b,
      /*c_mod=*/(short)0, c, /*reuse_a=*/false, /*reuse_b=*/false);
}

__device__ __forceinline__ float fast_rcp(float x) {
  return __builtin_amdgcn_rcpf(x);  // v_rcp_f32, avoids IEEE div chain
}

__device__ __forceinline__ float sigmoidf_(float x) {
  return fast_rcp(1.0f + __expf(-x));
}

// ---------------------------------------------------------------------------
// K1: q = grouped1x1(x, q_w); x_atten = x * sigmoid(q)
// one wave per (b, g, l-tile16); tiles = B*G*(L/16) = 4096
// ---------------------------------------------------------------------------
__global__ void k_qconv_sig(const float* __restrict__ x, const float* __restrict__ qw,
                            float* __restrict__ xatt) {
  int wid = blockIdx.x * (blockDim.x >> 5) + (threadIdx.x >> 5);
  int lane = threadIdx.x & 31;
  const int nlt = Ln / 16;
  if (wid >= Bn * Gn * nlt) return;
  int lt = wid % nlt; int t = wid / nlt; int g = t % Gn; int b = t / Gn;
  int l16 = lane & 15, hi = lane >> 4, k0 = 2 * hi;
  int l0 = lt * 16;

  const float* wrow = qw + ((size_t)(g * 16 + l16)) * 16;  // A: M = out-ch = l16
  v8f acc = {};
#pragma unroll
  for (int kc = 0; kc < 4; ++kc) {
    int k = 4 * kc + k0;
    v2f a, bb;
    a[0] = wrow[k];
    a[1] = wrow[k + 1];
    const float* xp = x + ((size_t)b * Cn + g * 16 + k) * Ln + l0 + l16;
    bb[0] = xp[0];
    bb[1] = xp[Ln];
    acc = wmma4(a, bb, acc);
  }
#pragma unroll
  for (int r = 0; r < 8; ++r) {
    int ch = g * 16 + r + 8 * hi;  // C: M = r + 8*hi, N = l16
    size_t idx = ((size_t)b * Cn + ch) * Ln + l0 + l16;
    xatt[idx] = x[idx] * sigmoidf_(acc[r]);
  }
}

// ---------------------------------------------------------------------------
// Generic grouped 1x1 conv (4 groups of 16->16): Y = W @ X per group
// ---------------------------------------------------------------------------
__global__ void k_conv1x1(const float* __restrict__ X, const float* __restrict__ W,
                          float* __restrict__ Y, int Llen) {
  int wid = blockIdx.x * (blockDim.x >> 5) + (threadIdx.x >> 5);
  int lane = threadIdx.x & 31;
  const int nlt = Llen / 16;
  if (wid >= Bn * Gn * nlt) return;
  int lt = wid % nlt; int t = wid / nlt; int g = t % Gn; int b = t / Gn;
  int l16 = lane & 15, hi = lane >> 4, k0 = 2 * hi;
  int l0 = lt * 16;

  const float* wrow = W + ((size_t)(g * 16 + l16)) * 16;
  v8f acc = {};
#pragma unroll
  for (int kc = 0; kc < 4; ++kc) {
    int k = 4 * kc + k0;
    v2f a, bb;
    a[0] = wrow[k];
    a[1] = wrow[k + 1];
    const float* xp = X + ((size_t)b * Cn + g * 16 + k) * Llen + l0 + l16;
    bb[0] = xp[0];
    bb[1] = xp[Llen];
    acc = wmma4(a, bb, acc);
  }
#pragma unroll
  for (int r = 0; r < 8; ++r) {
    int ch = g * 16 + r + 8 * hi;
    Y[((size_t)b * Cn + ch) * Llen + l0 + l16] = acc[r];
  }
}

// ---------------------------------------------------------------------------
// K3: xa = concat(odd(x_atten), even(x_atten)) over channels: (B,128,1024)
// ---------------------------------------------------------------------------
__global__ void k_split(const float* __restrict__ xatt, float* __restrict__ xa) {
  int idx = blockIdx.x * blockDim.x + threadIdx.x;
  if (idx >= Bn * 128 * Lh) return;
  int m = idx % Lh;
  int c2 = (idx / Lh) % 128;
  int b = idx / (128 * Lh);
  int cs = c2 & 63;
  int sl = (c2 < 64) ? (2 * m + 1) : (2 * m);
  xa[idx] = xatt[((size_t)b * Cn + cs) * Ln + sl];
}

// ---------------------------------------------------------------------------
// K4: down conv, K=3, groups=2 (in 64/group, out 32/group), pad 1 both sides.
// Implicit GEMM: Kdim = 192 = tap*64 + j.  waves = B*4*(Lh/16) = 2048
// ---------------------------------------------------------------------------
__global__ void k_downconv(const float* __restrict__ xa, const float* __restrict__ dw,
                           float* __restrict__ xd) {
  int wid = blockIdx.x * (blockDim.x >> 5) + (threadIdx.x >> 5);
  int lane = threadIdx.x & 31;
  const int nmt = Lh / 16;
  if (wid >= Bn * 4 * nmt) return;
  int mt = wid % nmt; int t = wid / nmt; int ct = t % 4; int b = t / 4;
  int l16 = lane & 15, hi = lane >> 4, k0 = 2 * hi;
  int c0 = ct * 16;
  int inb = 64 * (c0 >> 5);  // group = c/32 -> input base 64*g
  int m0 = mt * 16;
  int cc = c0 + l16;

  v8f acc = {};
#pragma unroll
  for (int kk = 0; kk < 48; ++kk) {
    int k = 4 * kk + k0;
    v2f a, bb;
#pragma unroll
    for (int u = 0; u < 2; ++u) {
      int kx = k + u;
      int j = kx & 63;
      int tap = kx >> 6;
      a[u] = dw[(size_t)cc * 192 + j * 3 + tap];
      int pos = m0 + l16 + tap - 1;
      bb[u] = (pos >= 0 && pos < Lh)
                  ? xa[((size_t)b * 128 + inb + j) * Lh + pos] : 0.0f;
    }
    acc = wmma4(a, bb, acc);
  }
#pragma unroll
  for (int r = 0; r < 8; ++r) {
    int ch = c0 + r + 8 * hi;
    xd[((size_t)b * Cn + ch) * Lh + m0 + l16] = acc[r];
  }
}

// ---------------------------------------------------------------------------
// Per-channel sum / sum-of-squares over (B, spatial).  One block per channel.
// float4 streaming (global_load_b128); groups never cross a channel row.
// ---------------------------------------------------------------------------
__global__ void k_stats(const float* __restrict__ X, float* __restrict__ sums, int Llen) {
  int c = blockIdx.x;
  int L4 = Llen >> 2;
  float s = 0.f, s2 = 0.f;
  for (int i = threadIdx.x; i < Bn * L4; i += blockDim.x) {
    int b = i / L4, l4 = i % L4;
    const float4* p = (const float4*)(X + ((size_t)b * Cn + c) * Llen) + l4;
    float4 v = *p;
    s += v.x + v.y + v.z + v.w;
    s2 += v.x * v.x + v.y * v.y + v.z * v.z + v.w * v.w;
  }
  __shared__ float sh[256], sh2[256];
  sh[threadIdx.x] = s;
  sh2[threadIdx.x] = s2;
  __syncthreads();
  for (int o = blockDim.x >> 1; o > 0; o >>= 1) {
    if ((int)threadIdx.x < o) {
      sh[threadIdx.x] += sh[threadIdx.x + o];
      sh2[threadIdx.x] += sh2[threadIdx.x + o];
    }
    __syncthreads();
  }
  if (threadIdx.x == 0) {
    sums[c] = sh[0];
    sums[64 + c] = sh2[0];
  }
}

// ---------------------------------------------------------------------------
// BN (batch statistics) + ReLU, elementwise, float4 vectorized.
// ---------------------------------------------------------------------------
__global__ void k_bnrelu(const float* __restrict__ X, const float* __restrict__ sums,
                         const float* __restrict__ gamma, const float* __restrict__ beta,
                         float* __restrict__ Y, int Llen) {
  int i4 = blockIdx.x * blockDim.x + threadIdx.x;
  int total4 = (Bn * Cn * Llen) >> 2;
  if (i4 >= total4) return;
  int c = (i4 / (Llen >> 2)) % Cn;
  float invN = fast_rcp((float)(Bn * Llen));
  float mean = sums[c] * invN;
  float var = sums[64 + c] * invN - mean * mean;
  float sc = rsqrtf(var + EPSc) * gamma[c];
  float off = beta[c] - mean * rsqrtf(var + EPSc) * gamma[c];
  float4 v = ((const float4*)X)[i4];
  float4 r;
  r.x = fmaxf(v.x * sc + off, 0.0f);
  r.y = fmaxf(v.y * sc + off, 0.0f);
  r.z = fmaxf(v.z * sc + off, 0.0f);
  r.w = fmaxf(v.w * sc + off, 0.0f);
  ((float4*)Y)[i4] = r;
}

// ---------------------------------------------------------------------------
// K9: flash attention per (b, h, l-tile16).  waves = B*H*(L/16) = 8192
// S^T = K^T @ Q (M=m, N=l)  -> softmax axis (m) lives across the 8 accum
// VGPRs + one shfl_xor(16); then O(e,l) += V @ P^T with pieces of P^T built
// from 8 precomputed shfl_xor(16) values (layout math per ISA 7.12.2).
// ---------------------------------------------------------------------------
__global__ void k_attn(const float* __restrict__ qbuf, const float* __restrict__ kbuf,
                       const float* __restrict__ vbuf, float* __restrict__ obuf) {
  int wid = blockIdx.x * (blockDim.x >> 5) + (threadIdx.x >> 5);
  int lane = threadIdx.x & 31;
  const int nlt = Ln / 16;
  if (wid >= Bn * Hn * nlt) return;
  int lt = wid % nlt; int t = wid / nlt; int h = t % Hn; int b = t / Hn;
  int l16 = lane & 15, hi = lane >> 4, k0 = 2 * hi;
  int l0 = lt * 16;
  const float qs = 0.35355339059327373f;  // 1/sqrt(E=8)

  // Q as B-operand: B[k=e][n=l]
  v2f qb[2];
#pragma unroll
  for (int ec = 0; ec < 2; ++ec) {
    int e = 4 * ec + k0;
    qb[ec][0] = qbuf[((size_t)b * Cn + 8 * h + e) * Ln + l0 + l16] * qs;
    qb[ec][1] = qbuf[((size_t)b * Cn + 8 * h + e + 1) * Ln + l0 + l16] * qs;
  }

  const float* krow = kbuf + ((size_t)b * Cn + 8 * h) * Lh;  // + e*Lh + m
  const float* vrow = vbuf + ((size_t)b * Cn + 8 * h) * Lh;

  v8f O = {};
  float rmax = -3.0e38f, rsum = 0.0f;

  for (int mt = 0; mt < Lh / 16; ++mt) {
    int m0 = mt * 16;
    // prefetch next K/V tile lines (speculative global_prefetch_b8)
    __builtin_prefetch(krow + (size_t)(4 * hi) * Lh + m0 + 16 + l16, 0, 0);
    __builtin_prefetch(vrow + (size_t)(4 * hi) * Lh + m0 + 16 + l16, 0, 0);

    // S^T tile: A = K^T (A[m][e]); lane: m = l16, e = 4ec + k0 + u
    v8f st = {};
#pragma unroll
    for (int ec = 0; ec < 2; ++ec) {
      int e = 4 * ec + k0;
      v2f ka;
      ka[0] = krow[(size_t)e * Lh + m0 + l16];
      ka[1] = krow[(size_t)(e + 1) * Lh + m0 + l16];
      st = wmma4(ka, qb[ec], st);
    }
    // online softmax over m (regs + partner half)
    float tmax = st[0];
#pragma unroll
    for (int r = 1; r < 8; ++r) tmax = fmaxf(tmax, st[r]);
    tmax = fmaxf(tmax, __shfl_xor(tmax, 16, 32));
    float nmax = fmaxf(rmax, tmax);
    float corr = __expf(rmax - nmax);
    float p[8];
    float ts = 0.0f;
#pragma unroll
    for (int r = 0; r < 8; ++r) {
      p[r] = __expf(st[r] - nmax);
      ts += p[r];
    }
    ts += __shfl_xor(ts, 16, 32);
    rsum = rsum * corr + ts;
    rmax = nmax;
#pragma unroll
    for (int r = 0; r < 8; ++r) O[r] *= corr;

    float px[8];
#pragma unroll
    for (int r = 0; r < 8; ++r) px[r] = __shfl_xor(p[r], 16, 32);

    // O(e,l) += V(e,m) * P^T(m,l) : A = V tile (rows e<8 valid), B = P^T
#pragma unroll
    for (int kc = 0; kc < 4; ++kc) {
      v2f va, pb;
#pragma unroll
      for (int u = 0; u < 2; ++u) {
        int m = 4 * kc + k0 + u;
        va[u] = (l16 < 8) ? vrow[(size_t)l16 * Lh + m0 + m] : 0.0f;
        int rlo = (4 * kc + u) & 7;
        int rhi = (4 * kc + 2 + u) & 7;
        // source lane for element (m,l) is l + 16*(m>=8); m>=8 iff kc>=2
        pb[u] = (kc < 2) ? (hi ? px[rhi] : p[rlo]) : (hi ? p[rhi] : px[rlo]);
      }
      O = wmma4(va, pb, O);
    }
  }

  float inv = fast_rcp(rsum);
#pragma unroll
  for (int r = 0; r < 8; ++r) O[r] *= inv;
  if (hi == 0) {
#pragma unroll
    for (int r = 0; r < 8; ++r)
      obuf[((size_t)b * Cn + 8 * h + r) * Ln + l0 + l16] = O[r];
  }
}

// ---------------------------------------------------------------------------
// K10: xs = out + x_atten; xc = conv(pad(cat[xs, q_, out]), out_w, G=4);
// xs2 = xc + xs.  Implicit GEMM, Kdim = 144 = tap*48 + j.
// ---------------------------------------------------------------------------
__device__ __forceinline__ float cat_fetch(const float* __restrict__ xatt,
                                           const float* __restrict__ qq,
                                           const float* __restrict__ ob,
                                           int b, int j, int l) {
  if ((unsigned)l >= (unsigned)Ln) return 0.0f;
  if (j < 64) {
    size_t idx = ((size_t)b * Cn + j) * Ln + l;
    return ob[idx] + xatt[idx];  // xs
  }
  if (j < 128) return qq[((size_t)b * Cn + (j - 64)) * Ln + l];
  return ob[((size_t)b * Cn + (j - 128)) * Ln + l];
}

__global__ void k_outconv(const float* __restrict__ xatt, const float* __restrict__ qq,
                          const float* __restrict__ ob, const float* __restrict__ ow,
                          float* __restrict__ xs2) {
  int wid = blockIdx.x * (blockDim.x >> 5) + (threadIdx.x >> 5);
  int lane = threadIdx.x & 31;
  const int nlt = Ln / 16;
  if (wid >= Bn * 4 * nlt) return;
  int lt = wid % nlt; int t = wid / nlt; int g = t % 4; int b = t / 4;
  int l16 = lane & 15, hi = lane >> 4, k0 = 2 * hi;
  int l0 = lt * 16;
  int cc = g * 16 + l16;

  v8f acc = {};
#pragma unroll
  for (int kk = 0; kk < 36; ++kk) {
    int k = 4 * kk + k0;
    v2f a, bb;
#pragma unroll
    for (int u = 0; u < 2; ++u) {
      int kx = k + u;
      int j = kx % 48;
      int tap = kx / 48;
      a[u] = ow[(size_t)cc * 144 + j * 3 + tap];
      int pos = l0 + l16 + tap - 1;
      bb[u] = cat_fetch(xatt, qq, ob, b, 48 * g + j, pos);
    }
    acc = wmma4(a, bb, acc);
  }
#pragma unroll
  for (int r = 0; r < 8; ++r) {
    int ch = g * 16 + r + 8 * hi;
    size_t idx = ((size_t)b * Cn + ch) * Ln + l0 + l16;
    float xs = ob[idx] + xatt[idx];
    xs2[idx] = acc[r] + xs;
  }
}

// ---------------------------------------------------------------------------
extern "C" void kernel_launch(void* const* d_in, const int* in_sizes, int n_in,
                              void* d_out, int out_size, void* d_ws, size_t ws_size,
                              hipStream_t stream) {
  const float* x          = (const float*)d_in[0];
  const float* q_w        = (const float*)d_in[1];
  const float* k_w        = (const float*)d_in[2];
  const float* v_w        = (const float*)d_in[3];
  const float* v1_w       = (const float*)d_in[4];
  const float* out_w      = (const float*)d_in[5];
  const float* down_w     = (const float*)d_in[6];
  const float* down_gamma = (const float*)d_in[7];
  const float* down_beta  = (const float*)d_in[8];
  const float* gamma      = (const float*)d_in[9];
  const float* beta       = (const float*)d_in[10];
  float* out = (float*)d_out;

  float* ws   = (float*)d_ws;
  float* xatt = ws;                 // B*C*L   = 1048576
  float* qq   = ws + 1048576;       // B*C*L
  float* xa   = ws + 2097152;       // B*128*Lh = 1048576
  float* xd   = ws + 3145728;       // B*C*Lh = 524288
  float* xab  = ws + 3670016;       // B*C*Lh
  float* vb   = ws + 4194304;       // B*C*Lh
  float* v1b  = ws + 4718592;       // B*C*Lh
  float* ob   = ws + 5242880;       // B*C*L
  float* xs2  = ws + 6291456;       // B*C*L
  float* st0  = ws + 7340032;       // 128 (down stats)
  float* st1  = ws + 7340160;       // 128 (final stats)

  // 1. q-conv + sigmoid gate
  k_qconv_sig<<<(Bn * Gn * (Ln / 16)) / 4, 128, 0, stream>>>(x, q_w, xatt);
  // 2. q_ = grouped1x1(x_atten, k_w)
  k_conv1x1<<<(Bn * Gn * (Ln / 16)) / 4, 128, 0, stream>>>(xatt, k_w, qq, Ln);
  // 3. odd/even channel split
  k_split<<<(Bn * 128 * Lh) / 256, 256, 0, stream>>>(xatt, xa);
  // 4. down conv (pre-BN)
  k_downconv<<<(Bn * 4 * (Lh / 16)) / 4, 128, 0, stream>>>(xa, down_w, xd);
  // 5./6. batchnorm (batch stats) + relu
  k_stats<<<64, 256, 0, stream>>>(xd, st0, Lh);
  k_bnrelu<<<(Bn * Cn * Lh / 4) / 256, 256, 0, stream>>>(xd, st0, down_gamma, down_beta, xab, Lh);
  // 7./8. v and v1 grouped 1x1 convs
  k_conv1x1<<<(Bn * Gn * (Lh / 16)) / 4, 128, 0, stream>>>(xab, v_w, vb, Lh);
  k_conv1x1<<<(Bn * Gn * (Lh / 16)) / 4, 128, 0, stream>>>(xab, v1_w, v1b, Lh);
  // 9. flash attention
  k_attn<<<(Bn * Hn * (Ln / 16)) / 4, 128, 0, stream>>>(qq, vb, v1b, ob);
  // 10. out conv + residuals
  k_outconv<<<(Bn * 4 * (Ln / 16)) / 4, 128, 0, stream>>>(xatt, qq, ob, out_w, xs2);
  // 11./12. final batchnorm + relu
  k_stats<<<64, 256, 0, stream>>>(xs2, st1, Ln);
  k_bnrelu<<<(Bn * Cn * Ln / 4) / 256, 256, 0, stream>>>(xs2, st1, gamma, beta, out, Ln);
}